// EmbeddingPooler_20572893347850
// MI455X (gfx1250) — compile-verified
//
#include <hip/hip_runtime.h>
#include <stdint.h>

#define VOCAB    100000
#define NROWS    (VOCAB + 2)             // 100002 table rows
#define DIM      128
#define BATCH    4096
#define MAX_LEN  200

#define CHUNK    8                       // tokens per TDM gather (32-bit index max)
#define WAVES    8                       // waves per block (256 threads)
#define ROWBYTES (DIM * 4)               // 512 B per embedding row
#define BUFBYTES (CHUNK * ROWBYTES)      // 4 KB per stage buffer
#define WAVE_LDS (2 * BUFBYTES)          // 8 KB double buffer per wave

typedef __attribute__((ext_vector_type(4))) unsigned int v4u;
typedef __attribute__((ext_vector_type(8))) int          v8i;
typedef __attribute__((ext_vector_type(4))) int          v4i;

// Issue one TDM gather of up to CHUNK embedding rows into LDS at lds_addr.
// D# gather mode, 32-bit indices, data_size=4B, row width 128 elements.
__device__ __forceinline__ void tdm_gather_rows(const float* __restrict__ emb,
                                                unsigned lds_addr, int n,
                                                int i0, int i1, int i2, int i3,
                                                int i4, int i5, int i6, int i7)
{
    const unsigned long long ga = (unsigned long long)(size_t)emb;

    v4u g0;
    g0.x = 0xC0000001u;                    // count=1 | gather_index_size=32b | gather_mode=1
    g0.y = lds_addr;                       // lds_addr[31:0]
    g0.z = (unsigned)(ga & 0xFFFFFFFFu);   // global_addr[31:0]
    g0.w = (unsigned)((ga >> 32) & 0x01FFFFFFu) | 0x80000000u; // addr[56:32] | type=2

    v8i g1;
    g1[0] = 0x00020000;                    // workgroup_mask=0, data_size=2 (4B)
    g1[1] = (int)(128u << 16);             // tensor_dim0[15:0]=128 @ bits 63:48
    g1[2] = (int)(0x86A2u << 16);          // tensor_dim0[31:16]=0, tensor_dim1 lo16=0x86A2
    g1[3] = (int)((128u << 16) | 0x1u);    // tensor_dim1 hi16=1 (100002), tile_dim0=128
    g1[4] = n;                             // tile_dim1 = #valid indices, tile_dim2=0
    g1[5] = 128;                           // tensor_dim0_stride[31:0] = 128 elements
    g1[6] = 0;                             // stride hi / dim1_stride (ignored in gather)
    g1[7] = 0;

    v4i g2 = { i0, i1, i2, i3 };           // 32-bit row indices 0..3
    v4i g3 = { i4, i5, i6, i7 };           // 32-bit row indices 4..7

#if defined(__clang_major__) && (__clang_major__ >= 23)
    v8i z8 = (v8i)0;
    __builtin_amdgcn_tensor_load_to_lds(g0, g1, g2, g3, z8, 0);
#else
    __builtin_amdgcn_tensor_load_to_lds(g0, g1, g2, g3, 0);
#endif
}

// Read 8 token ids for chunk starting at t0 (lanes 0..7 load, broadcast later).
__device__ __forceinline__ int load_ids(const int* __restrict__ tidp, int t0, int lane)
{
    int idx = t0 + (lane & (CHUNK - 1));
    if (idx > MAX_LEN - 1) idx = MAX_LEN - 1;     // clamp trailing chunk (no OOB)
    return tidp[idx];
}

__device__ __forceinline__ void issue_chunk(const float* __restrict__ emb,
                                            const int* __restrict__ tidp,
                                            int t0, int n,
                                            unsigned lds_buf_hw, int lane)
{
    int myid = load_ids(tidp, t0, lane);
    tdm_gather_rows(emb, lds_buf_hw, n,
                    __builtin_amdgcn_readlane(myid, 0),
                    __builtin_amdgcn_readlane(myid, 1),
                    __builtin_amdgcn_readlane(myid, 2),
                    __builtin_amdgcn_readlane(myid, 3),
                    __builtin_amdgcn_readlane(myid, 4),
                    __builtin_amdgcn_readlane(myid, 5),
                    __builtin_amdgcn_readlane(myid, 6),
                    __builtin_amdgcn_readlane(myid, 7));
}

__global__ __launch_bounds__(WAVES * 32)
void emb_pool_kernel(const float* __restrict__ emb,
                     const int* __restrict__ token_ids,
                     const int* __restrict__ lengths,
                     float* __restrict__ out)
{
    __shared__ __align__(16) unsigned char smem[WAVES * WAVE_LDS];   // 64 KB

    const int wave = threadIdx.x >> 5;
    const int lane = threadIdx.x & 31;
    const int row  = blockIdx.x * WAVES + wave;      // grid sized exactly

    // One row per wave -> length is wave-uniform; broadcast to SGPR so all
    // chunk bookkeeping becomes scalar control flow (s_cbranch, no exec masks).
    const int  len  = __builtin_amdgcn_readfirstlane(lengths[row]);  // 1..200
    const int* tidp = token_ids + (size_t)row * MAX_LEN;

    // Hardware (workgroup-relative) LDS byte offset of this wave's buffers.
    const unsigned lds_wave_hw =
        (unsigned)(size_t)(&smem[0]) + (unsigned)(wave * WAVE_LDS);
    const unsigned my_off = (unsigned)(wave * WAVE_LDS);

    float sx = 0.f, sy = 0.f, sz = 0.f, sw = 0.f;
    float mx = -INFINITY, my = -INFINITY, mz = -INFINITY, mw = -INFINITY;

    // ---- software-pipelined TDM gather loop (double buffered) ----
    int pend = (len < CHUNK) ? len : CHUNK;
    int t    = pend;
    int buf  = 0;
    issue_chunk(emb, tidp, 0, pend, lds_wave_hw, lane);

    while (pend > 0) {
        int n1 = len - t;
        if (n1 > CHUNK) n1 = CHUNK;
        if (n1 > 0) {
            issue_chunk(emb, tidp, t, n1,
                        lds_wave_hw + (unsigned)((buf ^ 1) * BUFBYTES), lane);
            // 2 tensor ops outstanding at most; in-order per wave, so <=1
            // remaining means the current chunk's DMA has landed in LDS.
            __builtin_amdgcn_s_wait_tensorcnt(1);
        } else {
            __builtin_amdgcn_s_wait_tensorcnt(0);
        }
        asm volatile("" ::: "memory");   // keep LDS reads below the wait

        const unsigned base = my_off + (unsigned)(buf * BUFBYTES) + (unsigned)(lane << 4);
#pragma unroll
        for (int s = 0; s < CHUNK; ++s) {
            if (s < pend) {              // scalar guard (pend is uniform)
                const float4 v =
                    *(const float4*)(const void*)&smem[base + s * ROWBYTES];
                sx += v.x; sy += v.y; sz += v.z; sw += v.w;
                mx = fmaxf(mx, v.x); my = fmaxf(my, v.y);
                mz = fmaxf(mz, v.z); mw = fmaxf(mw, v.w);
            }
        }
        buf ^= 1;
        pend = n1;
        t += n1;
    }

    // ---- epilogue: mean = sum * (1/len), concat [mean | max] ----
    const float inv = 1.0f / (float)len;
    float* orow = out + (size_t)row * (2 * DIM);
    float4 m4 = make_float4(sx * inv, sy * inv, sz * inv, sw * inv);
    float4 x4 = make_float4(mx, my, mz, mw);
    *(float4*)(orow + (lane << 2))        = m4;
    *(float4*)(orow + DIM + (lane << 2))  = x4;
}

extern "C" void kernel_launch(void* const* d_in, const int* in_sizes, int n_in,
                              void* d_out, int out_size, void* d_ws, size_t ws_size,
                              hipStream_t stream) {
    const float* emb  = (const float*)d_in[0];   // (100002, 128) f32
    const int*   tok  = (const int*)d_in[1];     // (4096, 200) i32
    const int*   lens = (const int*)d_in[2];     // (4096,) i32
    float*       out  = (float*)d_out;           // (4096, 256) f32

    dim3 grid(BATCH / WAVES);                    // 512 blocks x 256 threads
    dim3 block(WAVES * 32);
    emb_pool_kernel<<<grid, block, 0, stream>>>(emb, tok, lens, out);
}